// ModelGCNAttn_77884936945816
// MI455X (gfx1250) — compile-verified
//
#include <hip/hip_runtime.h>
#include <math.h>

#define B_   64
#define R_   148
#define D_   64
#define H_   4
#define HD_  16
#define INR_ 148
#define HID_ 1000
#define N1_  262144
#define E1_  1048576
#define N2_  (B_*R_)      /* 9472 */
#define E2_  262144
#define EPS_ 1e-5f

typedef __attribute__((ext_vector_type(2))) float v2f;
typedef __attribute__((ext_vector_type(8))) float v8f;

__device__ __forceinline__ void atomAddF(float* p, float v) {
  __hip_atomic_fetch_add(p, v, __ATOMIC_RELAXED, __HIP_MEMORY_SCOPE_AGENT);
}

// ---------------------------------------------------------------------------
// fp32 WMMA GEMM: C[M x N] = A[M x K] @ B[K x N] (+bias, optional relu).
// One wave computes a 32x64 tile: 2 A fragments x 4 B fragments ->
// 8 independent V_WMMA_F32_16X16X4_F32 per K-step.
// M multiple of 32 (all call sites: 262144, 9472, 64). K multiple of 4.
// GUARD=1 only when N is not a multiple of 64 (classifier, N=1000): column
// overflow handled by clamping to a legal address + 0/1 mask multiply, so
// loads stay unconditional. GUARD=0 drops all masking from the hot loop.
// TB: B stored as W[N x K] row-major, used transposed.
// All global accesses use a uniform base + 32-bit element offset so codegen
// can use SGPR-base + VGPR-offset global addressing (no flat, no 64-bit
// per-lane pointer arithmetic in the loop).
// ---------------------------------------------------------------------------
template <int TB, int GUARD>
__global__ __launch_bounds__(32) void k_gemm_wmma(
    const float* __restrict__ A, int lda,
    const float* __restrict__ Bm, int ldb,
    const float* __restrict__ bias,
    float* __restrict__ C, int ldc,
    int N, int K, int act) {
  const int n0 = blockIdx.x * 64;
  const int m0 = blockIdx.y * 32;
  const int lane = threadIdx.x;
  const int r  = lane & 15;
  const int hi = lane >> 4;            // 0: K pair {0,1}, 1: K pair {2,3}

  v8f acc[2][4];
#pragma unroll
  for (int i = 0; i < 2; ++i)
#pragma unroll
    for (int j = 0; j < 4; ++j)
      acc[i][j] = (v8f){0.f,0.f,0.f,0.f,0.f,0.f,0.f,0.f};

  unsigned aoff0 = (unsigned)(m0 + r) * (unsigned)lda + 2u * hi;
  unsigned aoff1 = aoff0 + 16u * (unsigned)lda;

  float msk[4];
  unsigned boff[4];
#pragma unroll
  for (int j = 0; j < 4; ++j) {
    const int n  = n0 + j * 16 + r;
    const int nc = GUARD ? ((n < N) ? n : (N - 1)) : n;
    msk[j] = (n < N) ? 1.f : 0.f;
    boff[j] = TB ? (unsigned)nc * (unsigned)ldb + 2u * hi
                 : 2u * hi * (unsigned)ldb + (unsigned)nc;
  }
  const unsigned bstep = TB ? 4u : 4u * (unsigned)ldb;
  const unsigned bld   = TB ? 1u : (unsigned)ldb;

  for (int k = 0; k < K; k += 4) {
    v2f a0, a1;
    a0.x = A[aoff0];
    a0.y = A[aoff0 + 1];
    a1.x = A[aoff1];
    a1.y = A[aoff1 + 1];
    aoff0 += 4;
    aoff1 += 4;
#pragma unroll
    for (int j = 0; j < 4; ++j) {
      v2f b;
      b.x = Bm[boff[j]];
      b.y = Bm[boff[j] + bld];
      if (GUARD) { b.x *= msk[j]; b.y *= msk[j]; }
      boff[j] += bstep;
      acc[0][j] = __builtin_amdgcn_wmma_f32_16x16x4_f32(false, a0, false, b,
                                                        (short)0, acc[0][j],
                                                        false, false);
      acc[1][j] = __builtin_amdgcn_wmma_f32_16x16x4_f32(false, a1, false, b,
                                                        (short)0, acc[1][j],
                                                        false, false);
    }
  }

#pragma unroll
  for (int i = 0; i < 2; ++i) {
#pragma unroll
    for (int j = 0; j < 4; ++j) {
      const int n = n0 + j * 16 + r;
      if (!GUARD || n < N) {
        const float bb = bias ? bias[n] : 0.f;
#pragma unroll
        for (int g = 0; g < 8; ++g) {
          float v = acc[i][j][g] + bb;
          if (act == 1) v = fmaxf(v, 0.f);
          C[(size_t)(m0 + i * 16 + g + 8 * hi) * ldc + n] = v;
        }
      }
    }
  }
}

// ---------------------------------------------------------------------------
__global__ void k_zero(float* __restrict__ p, size_t n) {
  size_t i = (size_t)blockIdx.x * blockDim.x + threadIdx.x;
  size_t step = (size_t)gridDim.x * blockDim.x;
  for (; i < n; i += step) p[i] = 0.f;
}

__global__ void k_deg(const int* __restrict__ dst, const float* __restrict__ ew,
                      float* __restrict__ deg, int E) {
  int e = blockIdx.x * blockDim.x + threadIdx.x;
  if (e < E) atomAddF(&deg[dst[e]], ew[e]);
}

__global__ void k_rsqrt_deg(float* __restrict__ d, int n) {
  int i = blockIdx.x * blockDim.x + threadIdx.x;
  if (i < n) d[i] = rsqrtf(d[i] + 1.0f);
}

// one block (64 threads) per edge: agg[dst] += h[src] * dinv[src]*ew*dinv[dst]
__global__ __launch_bounds__(64) void k_edge_scatter(
    const int* __restrict__ src, const int* __restrict__ dst,
    const float* __restrict__ ew, const float* __restrict__ dinv,
    const float* __restrict__ h, float* __restrict__ agg) {
  const int e = blockIdx.x;
  __shared__ float cf;
  __shared__ int ss, dd;
  if (threadIdx.x == 0) {
    ss = src[e];
    dd = dst[e];
    cf = dinv[ss] * ew[e] * dinv[dd];
  }
  __syncthreads();
  atomAddF(&agg[(size_t)dd * D_ + threadIdx.x],
           h[(size_t)ss * D_ + threadIdx.x] * cf);
}

// agg = relu(agg + h * dinv^2 + bias)   (self-loop + bias + relu)
__global__ __launch_bounds__(64) void k_gcn_finalize(
    float* __restrict__ agg, const float* __restrict__ h,
    const float* __restrict__ dinv, const float* __restrict__ bias) {
  const int i = blockIdx.x;
  const int d = threadIdx.x;
  const float di = dinv[i];
  const size_t o = (size_t)i * D_ + d;
  agg[o] = fmaxf(agg[o] + h[o] * di * di + bias[d], 0.f);
}

__global__ __launch_bounds__(64) void k_pool_accum(
    const float* __restrict__ h, const int* __restrict__ roi,
    const int* __restrict__ batch, float* __restrict__ sum,
    float* __restrict__ cnt) {
  const int i = blockIdx.x;
  const int seg = batch[i] * R_ + roi[i];
  atomAddF(&sum[(size_t)seg * D_ + threadIdx.x], h[(size_t)i * D_ + threadIdx.x]);
  if (threadIdx.x == 0) atomAddF(&cnt[seg], 1.f);
}

__global__ void k_pool_final(const float* __restrict__ sum,
                             const float* __restrict__ cnt,
                             float* __restrict__ pooled) {
  int idx = blockIdx.x * blockDim.x + threadIdx.x;
  if (idx >= N2_ * D_) return;
  pooled[idx] = sum[idx] / fmaxf(cnt[idx >> 6], 1.f);
}

__global__ void k_add2(const float* __restrict__ a, const float* __restrict__ b,
                       float* __restrict__ c, int n) {
  int i = blockIdx.x * blockDim.x + threadIdx.x;
  if (i < n) c[i] = a[i] + b[i];
}

// scores + softmax for one (b, h, qi) row; writes attn weights to d_out
__global__ __launch_bounds__(256) void k_attn_softmax(
    const float* __restrict__ qkv, float* __restrict__ attn) {
  const int bid = blockIdx.x;            // (b*H + h)*R + qi
  const int qi = bid % R_;
  const int h  = (bid / R_) % H_;
  const int b  = bid / (R_ * H_);
  __shared__ float qs[HD_];
  __shared__ float red[256];
  const int t = threadIdx.x;
  if (t < HD_) qs[t] = qkv[((size_t)(b * R_ + qi)) * (3 * D_) + h * HD_ + t];
  __syncthreads();
  float s = -3.0e38f;
  if (t < R_) {
    const float* kp = qkv + ((size_t)(b * R_ + t)) * (3 * D_) + D_ + h * HD_;
    float a = 0.f;
#pragma unroll
    for (int d2 = 0; d2 < HD_; ++d2) a += qs[d2] * kp[d2];
    s = a * 0.25f;                      // 1/sqrt(16)
  }
  red[t] = s;
  __syncthreads();
  for (int st = 128; st >= 1; st >>= 1) {
    if (t < st) red[t] = fmaxf(red[t], red[t + st]);
    __syncthreads();
  }
  const float mx = red[0];
  __syncthreads();
  const float e = (t < R_) ? __expf(s - mx) : 0.f;
  red[t] = e;
  __syncthreads();
  for (int st = 128; st >= 1; st >>= 1) {
    if (t < st) red[t] += red[t + st];
    __syncthreads();
  }
  const float inv = 1.f / red[0];
  if (t < R_) attn[(size_t)bid * R_ + t] = e * inv;
}

// o[b,qi,h,d] = sum_k attn[b,h,qi,k] * v[b,k,h,d]
__global__ void k_attn_apply(const float* __restrict__ attn,
                             const float* __restrict__ qkv,
                             float* __restrict__ o) {
  int idx = blockIdx.x * blockDim.x + threadIdx.x;  // (b*R+qi)*64 + h*16+d
  if (idx >= N2_ * D_) return;
  const int d  = idx & 15;
  const int h  = (idx >> 4) & 3;
  const int nq = idx >> 6;                          // b*R + qi
  const int b  = nq / R_;
  const float* aw = attn + (((size_t)(b * H_ + h)) * R_ + (nq % R_)) * R_;
  const float* vp = qkv + (size_t)b * R_ * (3 * D_) + 2 * D_ + h * HD_ + d;
  float a = 0.f;
  for (int kk = 0; kk < R_; ++kk) a += aw[kk] * vp[(size_t)kk * (3 * D_)];
  o[idx] = a;
}

// Out = LayerNorm(X + Y) with gain/bias; one 64-thread block per row
__global__ __launch_bounds__(64) void k_add_ln(
    const float* __restrict__ X, const float* __restrict__ Y,
    const float* __restrict__ g, const float* __restrict__ bt,
    float* __restrict__ Out) {
  const int row = blockIdx.x;
  const int d = threadIdx.x;
  __shared__ float sm[64];
  const float v = X[(size_t)row * 64 + d] + Y[(size_t)row * 64 + d];
  sm[d] = v;
  __syncthreads();
  for (int st = 32; st >= 1; st >>= 1) {
    if (d < st) sm[d] += sm[d + st];
    __syncthreads();
  }
  const float mu = sm[0] * (1.f / 64.f);
  __syncthreads();
  const float c = v - mu;
  sm[d] = c * c;
  __syncthreads();
  for (int st = 32; st >= 1; st >>= 1) {
    if (d < st) sm[d] += sm[d + st];
    __syncthreads();
  }
  const float var = sm[0] * (1.f / 64.f);
  Out[(size_t)row * 64 + d] = c * rsqrtf(var + EPS_) * g[d] + bt[d];
}

// eval-mode BatchNorm (mean 0, var 1) + LeakyReLU(0.01), z stored with ld=1008
__global__ void k_bn_leaky(float* __restrict__ z, const float* __restrict__ g,
                           const float* __restrict__ bb) {
  int idx = blockIdx.x * blockDim.x + threadIdx.x;
  if (idx >= B_ * HID_) return;
  const int b = idx / HID_, j = idx - b * HID_;
  float v = z[(size_t)b * 1008 + j];
  v = g[j] * v * rsqrtf(1.f + EPS_) + bb[j];
  z[(size_t)b * 1008 + j] = (v >= 0.f) ? v : 0.01f * v;
}

__global__ __launch_bounds__(128) void k_final_linear(
    const float* __restrict__ z, const float* __restrict__ W,
    const float* __restrict__ bb, float* __restrict__ out) {
  const int t = threadIdx.x;           // 128 = 64 graphs x 2 classes
  const int b = t >> 1, c = t & 1;
  float a = bb[c];
  const float* zr = z + (size_t)b * 1008;
  for (int k = 0; k < HID_; ++k) a += zr[k] * W[k * 2 + c];
  out[b * 2 + c] = a;
}

// ---------------------------------------------------------------------------
static inline void gemm(const float* A, int lda, const float* Bm, int ldb,
                        int transB, const float* bias, float* C, int ldc,
                        int M, int N, int K, int act, hipStream_t s) {
  dim3 g((unsigned)((N + 63) / 64), (unsigned)(M / 32));
  const bool guard = (N % 64) != 0;
  if (transB) {
    if (guard)
      k_gemm_wmma<1,1><<<g, 32, 0, s>>>(A, lda, Bm, ldb, bias, C, ldc, N, K, act);
    else
      k_gemm_wmma<1,0><<<g, 32, 0, s>>>(A, lda, Bm, ldb, bias, C, ldc, N, K, act);
  } else {
    if (guard)
      k_gemm_wmma<0,1><<<g, 32, 0, s>>>(A, lda, Bm, ldb, bias, C, ldc, N, K, act);
    else
      k_gemm_wmma<0,0><<<g, 32, 0, s>>>(A, lda, Bm, ldb, bias, C, ldc, N, K, act);
  }
}

extern "C" void kernel_launch(void* const* d_in, const int* in_sizes, int n_in,
                              void* d_out, int out_size, void* d_ws, size_t ws_size,
                              hipStream_t stream) {
  (void)in_sizes; (void)n_in; (void)out_size; (void)ws_size;
  const float* x_feat = (const float*)d_in[0];
  const int*   node_roi = (const int*)d_in[1];
  const int*   batch1 = (const int*)d_in[2];
  const int*   ei1 = (const int*)d_in[3];
  const float* ew1 = (const float*)d_in[4];
  const float* x2  = (const float*)d_in[5];
  const int*   roi2 = (const int*)d_in[6];
  const int*   batch2 = (const int*)d_in[7];
  const int*   ei2 = (const int*)d_in[8];
  const float* ew2 = (const float*)d_in[9];
  const float* Wg1 = (const float*)d_in[10]; const float* bg1 = (const float*)d_in[11];
  const float* Wg2 = (const float*)d_in[12]; const float* bg2 = (const float*)d_in[13];
  const float* Wr1 = (const float*)d_in[14]; const float* br1 = (const float*)d_in[15];
  const float* Wr2 = (const float*)d_in[16]; const float* br2 = (const float*)d_in[17];
  const float* ipw = (const float*)d_in[18]; const float* ipb = (const float*)d_in[19];
  const float* opw = (const float*)d_in[20]; const float* opb = (const float*)d_in[21];
  const float* ln1g = (const float*)d_in[22]; const float* ln1b = (const float*)d_in[23];
  const float* fw1 = (const float*)d_in[24]; const float* fb1 = (const float*)d_in[25];
  const float* fw2 = (const float*)d_in[26]; const float* fb2 = (const float*)d_in[27];
  const float* ln2g = (const float*)d_in[28]; const float* ln2b = (const float*)d_in[29];
  const float* Wc1 = (const float*)d_in[30]; const float* bc1 = (const float*)d_in[31];
  const float* bng = (const float*)d_in[32]; const float* bnb = (const float*)d_in[33];
  const float* Wc2 = (const float*)d_in[34]; const float* bc2 = (const float*)d_in[35];

  const int* s1 = ei1;  const int* dd1 = ei1 + E1_;
  const int* s2 = ei2;  const int* dd2 = ei2 + E2_;

  float* ws = (float*)d_ws;
  size_t o = 0;
  float* h1a   = ws + o; o += (size_t)N1_ * D_;
  float* h1b   = ws + o; o += (size_t)N1_ * D_;
  float* dinv1 = ws + o; o += N1_;
  float* h2a   = ws + o; o += (size_t)N2_ * D_;
  float* h2b   = ws + o; o += (size_t)N2_ * D_;
  float* dinv2 = ws + o; o += N2_;
  float* psum  = ws + o; o += (size_t)N2_ * D_;
  float* pcnt  = ws + o; o += N2_;
  float* comb  = ws + o; o += (size_t)N2_ * D_;
  float* qkvb  = ws + o; o += (size_t)N2_ * 3 * D_;
  float* obuf  = ws + o; o += (size_t)N2_ * D_;
  float* xbuf  = ws + o; o += (size_t)N2_ * D_;
  float* fbuf  = ws + o; o += (size_t)N2_ * D_;
  float* zbuf  = ws + o; o += (size_t)B_ * 1008;

  float* out        = (float*)d_out;
  float* out_logits = out;                                   // (64, 2)
  float* out_pooled = out + 128;                             // (64, 148, 64)
  float* out_proi   = out_pooled + (size_t)N2_ * D_;         // (64, 148, 64)
  float* out_tout   = out_proi + (size_t)N2_ * D_;           // (64, 148, 64)
  float* out_attn   = out_tout + (size_t)N2_ * D_;           // (64, 4, 148, 148)

  // ---- symmetric-norm degrees (deg = segsum(ew, dst) + 1; dinv = rsqrt) ----
  k_zero<<<1024, 256, 0, stream>>>(dinv1, (size_t)N1_);
  k_zero<<<64, 256, 0, stream>>>(dinv2, (size_t)N2_);
  k_deg<<<(E1_ + 255) / 256, 256, 0, stream>>>(dd1, ew1, dinv1, E1_);
  k_deg<<<(E2_ + 255) / 256, 256, 0, stream>>>(dd2, ew2, dinv2, E2_);
  k_rsqrt_deg<<<(N1_ + 255) / 256, 256, 0, stream>>>(dinv1, N1_);
  k_rsqrt_deg<<<(N2_ + 255) / 256, 256, 0, stream>>>(dinv2, N2_);

  // ---- ROI-graph branch: two GCN layers + pool ----
  gemm(x2, INR_, Wr1, D_, 0, nullptr, h2a, D_, N2_, D_, INR_, 0, stream);
  k_zero<<<256, 256, 0, stream>>>(h2b, (size_t)N2_ * D_);
  k_edge_scatter<<<E2_, 64, 0, stream>>>(s2, dd2, ew2, dinv2, h2a, h2b);
  k_gcn_finalize<<<N2_, 64, 0, stream>>>(h2b, h2a, dinv2, br1);
  gemm(h2b, D_, Wr2, D_, 0, nullptr, h2a, D_, N2_, D_, D_, 0, stream);
  k_zero<<<256, 256, 0, stream>>>(h2b, (size_t)N2_ * D_);
  k_edge_scatter<<<E2_, 64, 0, stream>>>(s2, dd2, ew2, dinv2, h2a, h2b);
  k_gcn_finalize<<<N2_, 64, 0, stream>>>(h2b, h2a, dinv2, br2);
  k_zero<<<256, 256, 0, stream>>>(psum, (size_t)N2_ * D_);
  k_zero<<<64, 256, 0, stream>>>(pcnt, (size_t)N2_);
  k_pool_accum<<<N2_, 64, 0, stream>>>(h2b, roi2, batch2, psum, pcnt);
  k_pool_final<<<(N2_ * D_ + 255) / 256, 256, 0, stream>>>(psum, pcnt, out_proi);

  // ---- node-graph branch: two GCN layers + pool ----
  gemm(x_feat, 64, Wg1, D_, 0, nullptr, h1a, D_, N1_, D_, 64, 0, stream);
  k_zero<<<4096, 256, 0, stream>>>(h1b, (size_t)N1_ * D_);
  k_edge_scatter<<<E1_, 64, 0, stream>>>(s1, dd1, ew1, dinv1, h1a, h1b);
  k_gcn_finalize<<<N1_, 64, 0, stream>>>(h1b, h1a, dinv1, bg1);
  gemm(h1b, D_, Wg2, D_, 0, nullptr, h1a, D_, N1_, D_, D_, 0, stream);
  k_zero<<<4096, 256, 0, stream>>>(h1b, (size_t)N1_ * D_);
  k_edge_scatter<<<E1_, 64, 0, stream>>>(s1, dd1, ew1, dinv1, h1a, h1b);
  k_gcn_finalize<<<N1_, 64, 0, stream>>>(h1b, h1a, dinv1, bg2);
  k_zero<<<256, 256, 0, stream>>>(psum, (size_t)N2_ * D_);
  k_zero<<<64, 256, 0, stream>>>(pcnt, (size_t)N2_);
  k_pool_accum<<<N1_, 64, 0, stream>>>(h1b, node_roi, batch1, psum, pcnt);
  k_pool_final<<<(N2_ * D_ + 255) / 256, 256, 0, stream>>>(psum, pcnt, out_pooled);

  // ---- transformer block ----
  k_add2<<<(N2_ * D_ + 255) / 256, 256, 0, stream>>>(out_pooled, out_proi, comb,
                                                     N2_ * D_);
  gemm(comb, D_, ipw, D_, 1, ipb, qkvb, 3 * D_, N2_, 3 * D_, D_, 0, stream);
  k_attn_softmax<<<B_ * H_ * R_, 256, 0, stream>>>(qkvb, out_attn);
  k_attn_apply<<<(N2_ * D_ + 255) / 256, 256, 0, stream>>>(out_attn, qkvb, obuf);
  gemm(obuf, D_, opw, D_, 1, opb, fbuf, D_, N2_, D_, D_, 0, stream);
  k_add_ln<<<N2_, 64, 0, stream>>>(comb, fbuf, ln1g, ln1b, xbuf);
  gemm(xbuf, D_, fw1, D_, 0, fb1, fbuf, D_, N2_, D_, D_, 1, stream);   // relu
  gemm(fbuf, D_, fw2, D_, 0, fb2, obuf, D_, N2_, D_, D_, 0, stream);
  k_add_ln<<<N2_, 64, 0, stream>>>(xbuf, obuf, ln2g, ln2b, out_tout);

  // ---- classifier ----
  gemm(out_tout, R_ * D_, Wc1, HID_, 0, bc1, zbuf, 1008, B_, HID_, R_ * D_, 0,
       stream);
  k_bn_leaky<<<(B_ * HID_ + 255) / 256, 256, 0, stream>>>(zbuf, bng, bnb);
  k_final_linear<<<1, 128, 0, stream>>>(zbuf, Wc2, bc2, out_logits);
}